// DSGRLayers_14972255993989
// MI455X (gfx1250) — compile-verified
//
#include <hip/hip_runtime.h>

#define DD 128
#define LB 50
#define NUU 8192
#define NII 8192

typedef float v2f __attribute__((ext_vector_type(2)));
typedef float v8f __attribute__((ext_vector_type(8)));

__device__ __forceinline__ v8f wmma_f32_k4(v2f a, v2f b, v8f c) {
    return __builtin_amdgcn_wmma_f32_16x16x4_f32(
        false, a, false, b, (short)0, c, false, false);
}

// Accumulate a 16x16 tile over one 128-wide K panel:
// Arow = &A[row*128] (this lane's M row), Wc = &W[kbase*128 + ncol].
__device__ __forceinline__ v8f gemm_panel128(const float* __restrict__ Arow,
                                             const float* __restrict__ Wc,
                                             int half, v8f acc) {
    #pragma unroll
    for (int k0 = 0; k0 < 128; k0 += 4) {
        const int ka = k0 + 2 * half;          // this lane's K pair {ka, ka+1}
        v2f a, b;
        a.x = Arow[ka];
        a.y = Arow[ka + 1];
        b.x = Wc[ka * 128];
        b.y = Wc[(ka + 1) * 128];
        acc = wmma_f32_k4(a, b, acc);
    }
    return acc;
}

// C[M x 128] = act( [A0 | A1] (M x K) @ W (K x 128) ), K in {128, 256}.
// A0/A1 are each M x 128 row-major; W is K x 128 row-major.
// One wave computes a 16x16 tile via V_WMMA_F32_16X16X4_F32; 8 waves cover N=128.
template <int K, bool TANH>
__global__ __launch_bounds__(256) void dsgr_gemm_wmma_f32(
    const float* __restrict__ A0, const float* __restrict__ A1,
    const float* __restrict__ W, float* __restrict__ C)
{
    const int wave  = threadIdx.x >> 5;
    const int lane  = threadIdx.x & 31;
    const int half  = lane >> 4;      // 0: lanes 0-15, 1: lanes 16-31
    const int lm    = lane & 15;
    const int mBase = blockIdx.x * 16;
    const int nBase = wave * 16;
    const int row   = mBase + lm;
    const int ncol  = nBase + lm;

    v8f acc = {};
    acc = gemm_panel128(A0 + row * 128, W + ncol, half, acc);
    if (K == 256)
        acc = gemm_panel128(A1 + row * 128, W + 128 * 128 + ncol, half, acc);

    #pragma unroll
    for (int i = 0; i < 8; i++) {
        const int r = mBase + i + half * 8;
        float v = acc[i];
        if (TANH) v = tanhf(v);
        C[r * 128 + ncol] = v;
    }
}

// One workgroup (128 threads) per node. Mailbox (50 x 128 f32 = 25.6KB) staged
// once in LDS (row stride 129 -> conflict-free row & column access), reused by
// all four passes. Ranks from O(L^2) counting (matches stable double-argsort).
__global__ __launch_bounds__(128) void dsgr_reduce_attn(
    const float* __restrict__ dstH,   // [N,128]  query features
    const float* __restrict__ srcH,   // [Nsrc,128] neighbor feature table
    const int*   __restrict__ nbr,    // [N,L]
    const int*   __restrict__ tim,    // [N,L]
    const float* __restrict__ emb,    // [L,128]  positional (query side)
    const float* __restrict__ embk,   // [L,128]  positional (value side)
    float* __restrict__ outLong,      // [N,128]
    float* __restrict__ outShort)     // [N,128]
{
    __shared__ float mail[LB][DD + 1];
    __shared__ float hsh[DD];
    __shared__ int   tsh[LB];
    __shared__ int   rsh[LB];
    __shared__ float esh[LB];
    __shared__ float wsh[LB];
    __shared__ int   lastSh;

    const int node = blockIdx.x;
    const int t    = threadIdx.x;
    const float scale = 0.08838834764831845f;  // 1/sqrt(128)

    // ---- stage: query row, times, mailbox gather (each row: 512B coalesced)
    hsh[t] = dstH[node * DD + t];
    if (t < LB) tsh[t] = tim[node * LB + t];
    for (int l = 0; l < LB; l++) {
        const int nb = nbr[node * LB + l];         // uniform -> scalar load
        mail[l][t] = srcH[nb * DD + t];
    }
    __syncthreads();

    // ---- ranks (stable ascending) + long-term logits
    if (t < LB) {
        const int myT = tsh[t];
        int cnt = 0;
        for (int j = 0; j < LB; j++) {
            const int tj = tsh[j];
            cnt += (tj < myT) || (tj == myT && j < t);
        }
        const int r = (LB - 1) - cnt;              // reorder = L - rank - 1
        rsh[t] = r;
        float e = 0.f;
        for (int d = 0; d < DD; d++)
            e += (emb[r * DD + d] + mail[t][d]) * hsh[d];
        esh[t] = e * scale;
    }
    if (t == 0) {
        int best = tsh[0], bi = 0;
        for (int j = 1; j < LB; j++)
            if (tsh[j] > best) { best = tsh[j]; bi = j; }
        lastSh = bi;
    }
    __syncthreads();

    // ---- softmax weights (long-term)
    if (t < LB) {
        float m = esh[0];
        for (int j = 1; j < LB; j++) m = fmaxf(m, esh[j]);
        wsh[t] = __expf(esh[t] - m);
    }
    __syncthreads();
    float S = 0.f;
    for (int j = 0; j < LB; j++) S += wsh[j];      // broadcast reads
    const float invS = 1.f / S;

    // ---- h_long: thread t owns channel d=t
    float acc = 0.f;
    for (int l = 0; l < LB; l++)
        acc += wsh[l] * (mail[l][t] + embk[rsh[l] * DD + t]);
    outLong[node * DD + t] = acc * invS;

    const int last = lastSh;
    __syncthreads();                               // done reading esh/wsh

    // ---- short-term logits vs most-recent neighbor
    if (t < LB) {
        float e1 = 0.f;
        for (int d = 0; d < DD; d++)
            e1 += mail[last][d] * mail[t][d];      // broadcast + conflict-free
        esh[t] = e1 * scale;
    }
    __syncthreads();
    if (t < LB) {
        float m = esh[0];
        for (int j = 1; j < LB; j++) m = fmaxf(m, esh[j]);
        wsh[t] = __expf(esh[t] - m);
    }
    __syncthreads();
    float S1 = 0.f;
    for (int j = 0; j < LB; j++) S1 += wsh[j];
    const float invS1 = 1.f / S1;

    float acc1 = 0.f;
    for (int l = 0; l < LB; l++)
        acc1 += wsh[l] * mail[l][t];
    outShort[node * DD + t] = acc1 * invS1;
}

extern "C" void kernel_launch(void* const* d_in, const int* in_sizes, int n_in,
                              void* d_out, int out_size, void* d_ws, size_t ws_size,
                              hipStream_t stream) {
    const float* user           = (const float*)d_in[0];
    const float* item           = (const float*)d_in[1];
    const float* W_user         = (const float*)d_in[2];
    const float* W_item         = (const float*)d_in[3];
    const float* agg_gate_user  = (const float*)d_in[4];
    const float* agg_gate_item  = (const float*)d_in[5];
    const float* update_user    = (const float*)d_in[6];
    const float* update_item    = (const float*)d_in[7];
    const float* user_date_emb  = (const float*)d_in[8];
    const float* user_date_embk = (const float*)d_in[9];
    const float* item_date_emb  = (const float*)d_in[10];
    const float* item_date_embk = (const float*)d_in[11];
    const int*   user_nbr       = (const int*)d_in[12];
    const int*   user_nbr_time  = (const int*)d_in[13];
    const int*   item_nbr       = (const int*)d_in[14];
    const int*   item_nbr_time  = (const int*)d_in[15];

    const int ND = NUU * DD;                  // 8192*128
    float* ws      = (float*)d_ws;
    float* uh      = ws + 0 * (size_t)ND;
    float* ih      = ws + 1 * (size_t)ND;
    float* u_long  = ws + 2 * (size_t)ND;
    float* u_short = ws + 3 * (size_t)ND;
    float* i_long  = ws + 4 * (size_t)ND;
    float* i_short = ws + 5 * (size_t)ND;
    float* u_agg   = ws + 6 * (size_t)ND;
    float* i_agg   = ws + 7 * (size_t)ND;

    float* out_user = (float*)d_out;
    float* out_item = out_user + ND;

    const dim3 gblk(256);
    const int  gemmGridU = NUU / 16;
    const int  gemmGridI = NII / 16;

    // 1) projected features: uh = user@W_user, ih = item@W_item
    dsgr_gemm_wmma_f32<128, false><<<gemmGridU, gblk, 0, stream>>>(user, user, W_user, uh);
    dsgr_gemm_wmma_f32<128, false><<<gemmGridI, gblk, 0, stream>>>(item, item, W_item, ih);

    // 2) attention reductions (user aggregates item mailboxes, and vice versa)
    dsgr_reduce_attn<<<NUU, 128, 0, stream>>>(uh, ih, user_nbr, user_nbr_time,
                                              user_date_emb, user_date_embk,
                                              u_long, u_short);
    dsgr_reduce_attn<<<NII, 128, 0, stream>>>(ih, uh, item_nbr, item_nbr_time,
                                              item_date_emb, item_date_embk,
                                              i_long, i_short);

    // 3) gate projections: agg = [h_long | h_short] @ gate_W   (K=256)
    dsgr_gemm_wmma_f32<256, false><<<gemmGridU, gblk, 0, stream>>>(u_long, u_short, agg_gate_user, u_agg);
    dsgr_gemm_wmma_f32<256, false><<<gemmGridI, gblk, 0, stream>>>(i_long, i_short, agg_gate_item, i_agg);

    // 4) update: out = tanh([agg | x] @ update_W)   (K=256)
    dsgr_gemm_wmma_f32<256, true><<<gemmGridU, gblk, 0, stream>>>(u_agg, user, update_user, out_user);
    dsgr_gemm_wmma_f32<256, true><<<gemmGridI, gblk, 0, stream>>>(i_agg, item, update_item, out_item);
}